// GraphSAGENet_54030688584326
// MI455X (gfx1250) — compile-verified
//
#include <hip/hip_runtime.h>

// ---------------------------------------------------------------------------
// GraphSAGE (2-layer, mean aggregation) for MI455X / gfx1250 (wave32, WMMA).
// Dims fixed by the reference: IN=128, HID=128, OUT=64.
// ---------------------------------------------------------------------------

typedef __attribute__((ext_vector_type(2))) float v2f;
typedef __attribute__((ext_vector_type(8))) float v8f;

static __device__ __forceinline__ v8f wmma_f32_16x16x4(v2f a, v2f b, v8f c) {
    // 8 args: (neg_a, A, neg_b, B, c_mod, C, reuse_a, reuse_b)
    return __builtin_amdgcn_wmma_f32_16x16x4_f32(
        false, a, false, b, (short)0, c, false, false);
}

// -------------------------- utility kernels --------------------------------

__global__ void zero_f32(float* __restrict__ p, long long n) {
    long long i = (long long)blockIdx.x * blockDim.x + threadIdx.x;
    if (i < n) p[i] = 0.0f;
}

__global__ void degree_kernel(const int* __restrict__ dst,
                              float* __restrict__ deg, int n_edges) {
    int e = blockIdx.x * blockDim.x + threadIdx.x;
    if (e < n_edges) atomicAdd(&deg[dst[e]], 1.0f);
}

// One wave (32 lanes) per edge; each lane moves 4 consecutive floats.
// Gather is a coalesced 512B burst per edge; scatter is 4 f32 atomics/lane
// that resolve in L2 (feature tables fit in the 192MB L2).
__global__ void scatter_add_128(const float* __restrict__ feat,
                                const int* __restrict__ src,
                                const int* __restrict__ dst,
                                float* __restrict__ summed, int n_edges) {
    long long gid = (long long)blockIdx.x * blockDim.x + threadIdx.x;
    int e = (int)(gid >> 5);
    if (e >= n_edges) return;
    int c = ((int)gid & 31) * 4;
    const float4 v = *(const float4*)(feat + (size_t)src[e] * 128 + c);
    float* o = summed + (size_t)dst[e] * 128 + c;
    atomicAdd(o + 0, v.x);
    atomicAdd(o + 1, v.y);
    atomicAdd(o + 2, v.z);
    atomicAdd(o + 3, v.w);
}

// ------------------------ fused SAGEConv GEMM ------------------------------
// out[m, n] = relu( (summed[m,:]*invdeg[m]) @ Wl[:,n] + bias[n]
//                   + self_feat[m,:] @ Wr[:,n] )
// K = N = 128. Block = 256 threads = 8 waves; wave w owns column tile w
// (16 cols), block owns a 16-row slab. V_WMMA_F32_16X16X4_F32, 64 MACs/wave.
// NOTE: self_feat/out intentionally NOT restrict (layer 2 runs in place:
// all A-reads complete before the post-barrier C-stores; blocks own
// disjoint row slabs).
__global__ void sage_gemm_128(const float* __restrict__ summed,
                              const float* __restrict__ deg,
                              const float* self_feat,
                              const float* __restrict__ Wl,
                              const float* __restrict__ Wr,
                              const float* __restrict__ bias,
                              float* out, int n_nodes) {
    const int lane = threadIdx.x & 31;
    const int wave = threadIdx.x >> 5;
    const int r = lane & 15;   // A: row-in-tile, B/C: col-in-tile
    const int g = lane >> 4;   // half-wave group
    const int row0 = blockIdx.x * 16;
    const int col0 = wave * 16;
    const int col  = col0 + r;

    int arow = row0 + r;                       // clamped (no EXEC divergence)
    if (arow >= n_nodes) arow = n_nodes - 1;
    const float invd = 1.0f / fmaxf(deg[arow], 1.0f);
    const float* aagg  = summed    + (size_t)arow * 128;
    const float* aself = self_feat + (size_t)arow * 128;

    v8f acc;
    const float bv = bias[col];
#pragma unroll
    for (int i = 0; i < 8; ++i) acc[i] = bv;   // bias: same N for all 8 elems

    // mean-neighbor path: A = summed * invdeg, B = Wl
#pragma unroll 4
    for (int k = 0; k < 128; k += 4) {
        const int kb = k + 2 * g;
        v2f a; a.x = aagg[kb] * invd; a.y = aagg[kb + 1] * invd;
        v2f b; b.x = Wl[(size_t)kb * 128 + col];
               b.y = Wl[(size_t)(kb + 1) * 128 + col];
        acc = wmma_f32_16x16x4(a, b, acc);
    }
    // self path: A = self_feat, B = Wr
#pragma unroll 4
    for (int k = 0; k < 128; k += 4) {
        const int kb = k + 2 * g;
        v2f a; a.x = aself[kb]; a.y = aself[kb + 1];
        v2f b; b.x = Wr[(size_t)kb * 128 + col];
               b.y = Wr[(size_t)(kb + 1) * 128 + col];
        acc = wmma_f32_16x16x4(a, b, acc);
    }

#pragma unroll
    for (int i = 0; i < 8; ++i) acc[i] = fmaxf(acc[i], 0.0f);  // ReLU

    __syncthreads();  // all waves done reading A rows -> in-place store safe

#pragma unroll
    for (int i = 0; i < 8; ++i) {
        const int m = row0 + i + 8 * g;        // C/D layout: M = i + 8*g
        if (m < n_nodes) out[(size_t)m * 128 + col] = acc[i];
    }
}

// --------------------------- final linear (N=64) ---------------------------
// out = emb @ W_lin + b_lin  (no relu). Block = 128 threads = 4 waves.
__global__ void final_gemm_64(const float* __restrict__ A,
                              const float* __restrict__ W,
                              const float* __restrict__ bias,
                              float* __restrict__ out, int n_nodes) {
    const int lane = threadIdx.x & 31;
    const int wave = threadIdx.x >> 5;
    const int r = lane & 15;
    const int g = lane >> 4;
    const int row0 = blockIdx.x * 16;
    const int col  = wave * 16 + r;

    int arow = row0 + r;
    if (arow >= n_nodes) arow = n_nodes - 1;
    const float* ar = A + (size_t)arow * 128;

    v8f acc;
    const float bv = bias[col];
#pragma unroll
    for (int i = 0; i < 8; ++i) acc[i] = bv;

#pragma unroll 4
    for (int k = 0; k < 128; k += 4) {
        const int kb = k + 2 * g;
        v2f a; a.x = ar[kb]; a.y = ar[kb + 1];
        v2f b; b.x = W[(size_t)kb * 64 + col];
               b.y = W[(size_t)(kb + 1) * 64 + col];
        acc = wmma_f32_16x16x4(a, b, acc);
    }

#pragma unroll
    for (int i = 0; i < 8; ++i) {
        const int m = row0 + i + 8 * g;
        if (m < n_nodes) out[(size_t)m * 64 + col] = acc[i];
    }
}

// ------------------------------- launcher ----------------------------------

extern "C" void kernel_launch(void* const* d_in, const int* in_sizes, int n_in,
                              void* d_out, int out_size, void* d_ws, size_t ws_size,
                              hipStream_t stream) {
    const float* x     = (const float*)d_in[0];
    const int*   edges = (const int*)d_in[1];
    const float* W1_l  = (const float*)d_in[2];
    const float* b1    = (const float*)d_in[3];
    const float* W1_r  = (const float*)d_in[4];
    const float* W2_l  = (const float*)d_in[5];
    const float* b2    = (const float*)d_in[6];
    const float* W2_r  = (const float*)d_in[7];
    const float* W_lin = (const float*)d_in[8];
    const float* b_lin = (const float*)d_in[9];

    const int n_nodes = in_sizes[0] / 128;
    const int n_edges = in_sizes[1] / 2;
    const int* src = edges;
    const int* dst = edges + n_edges;

    // workspace: deg [n_nodes] | summed [n_nodes*128]  (~52 MB)
    float* deg    = (float*)d_ws;
    float* summed = deg + (1 << 17);   // 512 KB offset, aligned

    float* out_ptr = (float*)d_out;                         // [n_nodes, 64]
    float* emb_ptr = out_ptr + (size_t)n_nodes * 64;        // [n_nodes,128]
    // emb region doubles as storage for layer-1 activations h.

    const long long nn128 = (long long)n_nodes * 128;
    const long long sthreads = (long long)n_edges * 32;

    dim3 b256(256), b128(128);
    const unsigned zb_deg = (unsigned)((n_nodes + 255) / 256);
    const unsigned zb_sum = (unsigned)((nn128 + 255) / 256);
    const unsigned eb     = (unsigned)((n_edges + 255) / 256);
    const unsigned sb     = (unsigned)((sthreads + 255) / 256);
    const unsigned gb     = (unsigned)((n_nodes + 15) / 16);

    // degree (shared by both layers)
    zero_f32<<<zb_deg, b256, 0, stream>>>(deg, n_nodes);
    degree_kernel<<<eb, b256, 0, stream>>>(dst, deg, n_edges);

    // ---- layer 1 ----
    zero_f32<<<zb_sum, b256, 0, stream>>>(summed, nn128);
    scatter_add_128<<<sb, b256, 0, stream>>>(x, src, dst, summed, n_edges);
    sage_gemm_128<<<gb, b256, 0, stream>>>(summed, deg, x, W1_l, W1_r, b1,
                                           emb_ptr /*h*/, n_nodes);

    // ---- layer 2 (h -> emb, in place) ----
    zero_f32<<<zb_sum, b256, 0, stream>>>(summed, nn128);
    scatter_add_128<<<sb, b256, 0, stream>>>(emb_ptr /*h*/, src, dst, summed,
                                             n_edges);
    sage_gemm_128<<<gb, b256, 0, stream>>>(summed, deg, emb_ptr /*h*/, W2_l,
                                           W2_r, b2, emb_ptr /*emb*/, n_nodes);

    // ---- final linear ----
    final_gemm_64<<<gb, b128, 0, stream>>>(emb_ptr, W_lin, b_lin, out_ptr,
                                           n_nodes);
}